// LinePooling_412316860464
// MI455X (gfx1250) — compile-verified
//
#include <hip/hip_runtime.h>

#define TOKS_PER_BLOCK 16
#define MAX_NCH 256   // supports D up to 1024 (D=768 -> nch=192)

#if __has_builtin(__builtin_amdgcn_global_load_async_to_lds_b128)
#define LP_HAVE_ASYNC_LDS 1
#endif

typedef int lp_v4i __attribute__((ext_vector_type(4)));
typedef __attribute__((address_space(1))) lp_v4i* lp_gptr;   // global
typedef __attribute__((address_space(3))) lp_v4i* lp_lptr;   // LDS

__device__ __forceinline__ void lp_wait_async_le1() {
#if __has_builtin(__builtin_amdgcn_s_wait_asynccnt)
    __builtin_amdgcn_s_wait_asynccnt(1);
#else
    asm volatile("s_wait_asynccnt 0x1" ::: "memory");
#endif
}
__device__ __forceinline__ void lp_wait_async_le0() {
#if __has_builtin(__builtin_amdgcn_s_wait_asynccnt)
    __builtin_amdgcn_s_wait_asynccnt(0);
#else
    asm volatile("s_wait_asynccnt 0x0" ::: "memory");
#endif
}

// ---------------------------------------------------------------------------
// Scatter-add kernel. blockDim.x == D/4 threads (768 -> 192 = 6 wave32s).
// Each block: TOKS_PER_BLOCK tokens. Rows stream HBM -> LDS via
// GLOBAL_LOAD_ASYNC_TO_LDS_B128 (ASYNCcnt double-buffer, issued before the
// ids prologue so the first DMA overlaps it); accumulation via fp32 global
// atomics resolved in the 192MB L2.
// Thread tid owns stage[*][tid] exclusively (tid unique within the block),
// so only same-wave ASYNCcnt ordering is needed -- no barriers in the loop.
// ---------------------------------------------------------------------------
__global__ void lp_scatter(const float* __restrict__ h,
                           const int*   __restrict__ ids,
                           float* __restrict__ feat,   // [L, D]
                           float* __restrict__ cnt,    // [L] (mask region reused)
                           int T, int D) {
    __shared__ int sid[TOKS_PER_BLOCK];
#ifdef LP_HAVE_ASYNC_LDS
    __shared__ float4 stage[2][MAX_NCH];
#endif

    const int tid  = threadIdx.x;           // 0 .. D/4-1
    const int tok0 = blockIdx.x * TOKS_PER_BLOCK;
    const int nch  = D >> 2;                // float4 chunks per row
    int ntok = T - tok0;
    if (ntok > TOKS_PER_BLOCK) ntok = TOKS_PER_BLOCK;

    const float4* __restrict__ hv =
        reinterpret_cast<const float4*>(h) + (size_t)tok0 * nch;

#ifdef LP_HAVE_ASYNC_LDS
    // Prime token 0 -> buffer 0 immediately: DMA overlaps the ids prologue.
    __builtin_amdgcn_global_load_async_to_lds_b128(
        (lp_gptr)(const void*)&hv[tid],
        (lp_lptr)(void*)&stage[0][tid], 0, 0);
#endif

    if (tid < TOKS_PER_BLOCK) {
        int t  = tok0 + tid;
        int id = (t < T) ? ids[t] : -1;
        sid[tid] = id;
        if (id >= 0) {
            unsafeAtomicAdd(&cnt[id], 1.0f);   // global_atomic_add_f32 (no rtn)
        }
    }
    __syncthreads();

#ifdef LP_HAVE_ASYNC_LDS
    for (int t = 0; t < ntok; ++t) {
        if (t + 1 < ntok) {
            __builtin_amdgcn_global_load_async_to_lds_b128(
                (lp_gptr)(const void*)&hv[(size_t)(t + 1) * nch + tid],
                (lp_lptr)(void*)&stage[(t + 1) & 1][tid], 0, 0);
            lp_wait_async_le1();   // oldest (token t) transfer complete
        } else {
            lp_wait_async_le0();
        }
        float4 v = stage[t & 1][tid];
        int id = sid[t];
        if (id >= 0) {
            float* dst = feat + (size_t)id * D + (tid << 2);
            unsafeAtomicAdd(dst + 0, v.x);
            unsafeAtomicAdd(dst + 1, v.y);
            unsafeAtomicAdd(dst + 2, v.z);
            unsafeAtomicAdd(dst + 3, v.w);
        }
    }
#else
    #pragma unroll 4
    for (int t = 0; t < ntok; ++t) {
        if (t + 2 < ntok) {
            __builtin_prefetch(&hv[(size_t)(t + 2) * nch + tid], 0, 1);
        }
        float4 v = hv[(size_t)t * nch + tid];
        int id = sid[t];
        if (id >= 0) {
            float* dst = feat + (size_t)id * D + (tid << 2);
            unsafeAtomicAdd(dst + 0, v.x);
            unsafeAtomicAdd(dst + 1, v.y);
            unsafeAtomicAdd(dst + 2, v.z);
            unsafeAtomicAdd(dst + 3, v.w);
        }
    }
#endif
}

// ---------------------------------------------------------------------------
// Zero the whole output buffer (features + count/mask region) every call.
// ---------------------------------------------------------------------------
__global__ void lp_zero(float* __restrict__ p, int n4) {
    int i = blockIdx.x * blockDim.x + threadIdx.x;
    const float4 z = make_float4(0.f, 0.f, 0.f, 0.f);
    for (; i < n4; i += gridDim.x * blockDim.x) {
        reinterpret_cast<float4*>(p)[i] = z;
    }
}

// ---------------------------------------------------------------------------
// Finalize: one block per line; features /= max(count,1); count -> 0/1 mask.
// ---------------------------------------------------------------------------
__global__ void lp_finalize(float* __restrict__ feat,
                            float* __restrict__ cntmask,
                            int D) {
    const int l   = blockIdx.x;
    const int tid = threadIdx.x;            // 0 .. D/4-1

    const float c = cntmask[l];
    __syncthreads();   // everyone reads c before thread 0 overwrites it

    const float inv = 1.0f / fmaxf(c, 1.0f);

    float4* row = reinterpret_cast<float4*>(feat) + (size_t)l * (D >> 2);
    float4 v = row[tid];
    v.x *= inv; v.y *= inv; v.z *= inv; v.w *= inv;
    row[tid] = v;

    if (tid == 0) {
        cntmask[l] = (c > 0.0f) ? 1.0f : 0.0f;
    }
}

// ---------------------------------------------------------------------------
// Launch. Inputs: [hidden_states f32 (B*S*D), line_ids i32 (B*S),
// num_lines i32 scalar (unused; L derived from out_size)].
// d_out layout: [L*D feature floats][L mask floats].
// ---------------------------------------------------------------------------
extern "C" void kernel_launch(void* const* d_in, const int* in_sizes, int n_in,
                              void* d_out, int out_size, void* d_ws, size_t ws_size,
                              hipStream_t stream) {
    const float* h   = (const float*)d_in[0];
    const int*   ids = (const int*)d_in[1];

    const int T = in_sizes[1];                 // 65536 tokens
    const int D = in_sizes[0] / in_sizes[1];   // 768
    const int L = out_size / (D + 1);          // 1024 lines

    float* feat = (float*)d_out;               // [L, D]
    float* cnt  = feat + (size_t)L * D;        // [L] counts -> mask

    // 1) zero accumulators (features + counts)
    {
        int n4 = out_size >> 2;
        int blocks = (n4 + 1023) / 1024;
        if (blocks > 2048) blocks = 2048;
        lp_zero<<<blocks, 256, 0, stream>>>((float*)d_out, n4);
    }

    // 2) scatter-add
    {
        int blocks = (T + TOKS_PER_BLOCK - 1) / TOKS_PER_BLOCK;  // 4096
        lp_scatter<<<blocks, D / 4, 0, stream>>>(h, ids, feat, cnt, T, D);
    }

    // 3) finalize mean + mask
    {
        lp_finalize<<<L, D / 4, 0, stream>>>(feat, cnt, D);
    }
}